// GNN_20512763806291
// MI455X (gfx1250) — compile-verified
//
#include <hip/hip_runtime.h>
#include <hip/hip_bf16.h>

typedef __attribute__((ext_vector_type(2))) float v2f;
typedef __attribute__((ext_vector_type(8))) float v8f;

#define D 128

// ---------------------------------------------------------------------------
// Degree count: one thread per edge, float atomic on cnt[dst].
// ---------------------------------------------------------------------------
__global__ __launch_bounds__(256) void sage_count(const long long* __restrict__ dst,
                                                  float* __restrict__ cnt,
                                                  int n_edges) {
  int e = blockIdx.x * blockDim.x + threadIdx.x;
  if (e < n_edges) {
    atomicAdd(&cnt[(int)dst[e]], 1.0f);
  }
}

// ---------------------------------------------------------------------------
// Scatter-add of messages: thread t of a 128-thread group owns column (t&127)
// of one edge; grid-stride over E*128 elements. Gathers hit L2 (x fits in the
// 192MB L2), scatters are global_atomic_add_f32.
// ---------------------------------------------------------------------------
__global__ __launch_bounds__(256) void sage_scatter(const long long* __restrict__ src,
                                                    const long long* __restrict__ dst,
                                                    const float* __restrict__ x,
                                                    float* __restrict__ agg,
                                                    int n_edges) {
  const size_t total = (size_t)n_edges * D;
  const size_t stride = (size_t)gridDim.x * blockDim.x;
  for (size_t i = (size_t)blockIdx.x * blockDim.x + threadIdx.x; i < total; i += stride) {
    const int e = (int)(i >> 7);        // edge id
    const int t = (int)(i & (D - 1));   // column
    const int s = (int)src[e];
    const int d = (int)dst[e];
    atomicAdd(&agg[(size_t)d * D + t], x[(size_t)s * D + t]);
  }
}

// ---------------------------------------------------------------------------
// Fused:  out = (agg * 1/max(cnt,1)) @ Wl^T + bl + xin @ Wr^T   [+ ReLU]
// One wave -> one 16x16 tile. blockDim 256 = 8 waves covering the 8 N-tiles
// of the 128-wide output; blockIdx.x selects the 16-row M-tile.
//
// V_WMMA_F32_16X16X4_F32 layouts (wave32):
//   A 16x4 : lane<16 holds K={0,1}, lane>=16 holds K={2,3}, row M = lane&15
//   B 4x16 : lane<16 holds K={0,1}, lane>=16 holds K={2,3}, col N = lane&15
//   C/D    : VGPR v -> M = v + 8*(lane>=16), N = lane&15
// ---------------------------------------------------------------------------
__global__ __launch_bounds__(256) void sage_gemm(const float* __restrict__ agg,
                                                 const float* __restrict__ cnt,
                                                 const float* __restrict__ xin,
                                                 const float* __restrict__ Wl,
                                                 const float* __restrict__ bl,
                                                 const float* __restrict__ Wr,
                                                 float* __restrict__ out,
                                                 int n_nodes, int relu) {
  const int wave = threadIdx.x >> 5;   // 0..7 -> N tile
  const int lane = threadIdx.x & 31;
  const int half = lane >> 4;          // 0: K base +0, 1: K base +2
  const int l15  = lane & 15;
  const int m0   = blockIdx.x * 16;
  const int n0   = wave * 16;

  const int row  = m0 + l15;
  const int arow = (row < n_nodes) ? row : 0;   // keep EXEC full for WMMA
  const int col  = n0 + l15;

  const float scale = 1.0f / fmaxf(cnt[arow], 1.0f);

  const float* __restrict__ aP = agg + (size_t)arow * D + 2 * half;
  const float* __restrict__ xP = xin + (size_t)arow * D + 2 * half;
  const float* __restrict__ blP = Wl + (size_t)col * D + 2 * half;  // B[k][col]=Wl[col*D+k]
  const float* __restrict__ brP = Wr + (size_t)col * D + 2 * half;

  v8f acc = {};

  // lin_l: mean @ Wl^T  (mean division fused into the A-operand load)
#pragma unroll 8
  for (int k = 0; k < D; k += 4) {
    v2f a = *(const v2f*)(aP + k);
    a[0] *= scale;
    a[1] *= scale;
    v2f b = *(const v2f*)(blP + k);
    acc = __builtin_amdgcn_wmma_f32_16x16x4_f32(false, a, false, b,
                                                (short)0, acc, false, false);
  }

  // lin_r: x @ Wr^T
#pragma unroll 8
  for (int k = 0; k < D; k += 4) {
    v2f a = *(const v2f*)(xP + k);
    v2f b = *(const v2f*)(brP + k);
    acc = __builtin_amdgcn_wmma_f32_16x16x4_f32(false, a, false, b,
                                                (short)0, acc, false, false);
  }

  const float bias = bl[n0 + l15];
#pragma unroll
  for (int v = 0; v < 8; ++v) {
    const int m = m0 + v + 8 * half;
    if (m < n_nodes) {
      float val = acc[v] + bias;
      if (relu) val = fmaxf(val, 0.0f);
      out[(size_t)m * D + n0 + l15] = val;
    }
  }
}

// ---------------------------------------------------------------------------
// Host-side orchestration (graph-capture safe: only async ops on `stream`).
// ---------------------------------------------------------------------------
extern "C" void kernel_launch(void* const* d_in, const int* in_sizes, int n_in,
                              void* d_out, int out_size, void* d_ws, size_t ws_size,
                              hipStream_t stream) {
  const float*     x    = (const float*)d_in[0];
  const long long* ei   = (const long long*)d_in[1];   // int64 edge_index [2, E]
  const float*     Wl1  = (const float*)d_in[2];
  const float*     bl1  = (const float*)d_in[3];
  const float*     Wr1  = (const float*)d_in[4];
  const float*     Wl2  = (const float*)d_in[5];
  const float*     bl2  = (const float*)d_in[6];
  const float*     Wr2  = (const float*)d_in[7];
  const float*     Wl3  = (const float*)d_in[8];
  const float*     bl3  = (const float*)d_in[9];
  const float*     Wr3  = (const float*)d_in[10];

  const int n_nodes = in_sizes[0] / D;   // 50000
  const int n_edges = in_sizes[1] / 2;   // 800000

  const long long* src = ei;
  const long long* dst = ei + n_edges;

  // workspace layout: agg[N*D] | cnt[N] | h1[N*D] | h2[N*D]
  float* agg = (float*)d_ws;
  float* cnt = agg + (size_t)n_nodes * D;
  float* h1  = cnt + n_nodes;
  float* h2  = h1 + (size_t)n_nodes * D;
  float* out = (float*)d_out;

  const int mtiles = (n_nodes + 15) / 16;      // 3125
  const int cblk   = (n_edges + 255) / 256;

  auto layer = [&](const float* xin, float* xout, const float* Wl,
                   const float* blv, const float* Wr, int relu) {
    hipMemsetAsync(agg, 0, ((size_t)n_nodes * D + n_nodes) * sizeof(float), stream);
    sage_count<<<cblk, 256, 0, stream>>>(dst, cnt, n_edges);
    sage_scatter<<<8192, 256, 0, stream>>>(src, dst, xin, agg, n_edges);
    sage_gemm<<<mtiles, 256, 0, stream>>>(agg, cnt, xin, Wl, blv, Wr, xout, n_nodes, relu);
  };

  layer(x,  h1,  Wl1, bl1, Wr1, 1);   // layer 1 + ReLU
  layer(h1, h2,  Wl2, bl2, Wr2, 0);   // layer 2 (dropout = identity)
  layer(h2, out, Wl3, bl3, Wr3, 0);   // layer 3
}